// Model_37056977830318
// MI455X (gfx1250) — compile-verified
//
#include <hip/hip_runtime.h>
#include <math.h>

// ---------------------------------------------------------------------------
// Bidirectional LSTM (B=64, T=4096, E=H=15) + masked linear(30 -> 45).
// Strategy:
//   Kernel 1 (grid=2, one block per direction, 256 thr = 8 wave32):
//     sequential scan; per step a [64x32]@[32x64] f32 WMMA GEMM
//     (x|h concat, K padded to 32; gates padded to 64) + gate elementwise.
//     Writes hidden states into Hcat[T*B][32] (fwd cols 0..14, bwd 15..29,
//     bwd stored un-reversed at t = L-1-s).
//   Kernel 2 (16384 blocks x 128 thr): per 16-row tile, [16x32]@[32x48]
//     f32 WMMA with bias in the accumulator; invalid rows (t >= length)
//     zeroed at LDS staging so output there is exactly the bias.
// ---------------------------------------------------------------------------

typedef __attribute__((ext_vector_type(2))) float v2f;
typedef __attribute__((ext_vector_type(8))) float v8f;

#define TMAX 4096
#define BATCH 64
#define EDIM 15
#define HDIM 15
#define GDIM 60     // 4*H
#define KCOMB 32    // padded E(->16) + H(->16)
#define NPAD 64     // padded 4H

__device__ __forceinline__ float sigm_(float x) { return 1.0f / (1.0f + __expf(-x)); }

__global__ __launch_bounds__(256)
void lstm_seq_kernel(const int* __restrict__ data,      // (B,T) int32
                     const int* __restrict__ lengths,   // (B)   int32
                     const float* __restrict__ emb,     // (V,E)
                     const float* __restrict__ h0,      // (2,B,H)
                     const float* __restrict__ c0,      // (2,B,H)
                     const float* __restrict__ w_ih_f,  // (4H,E)
                     const float* __restrict__ w_hh_f,  // (4H,H)
                     const float* __restrict__ b_f,     // (4H)
                     const float* __restrict__ w_ih_b,
                     const float* __restrict__ w_hh_b,
                     const float* __restrict__ b_b,
                     float* __restrict__ Hcat)          // (T*B, 32)
{
    __shared__ float wcomb[KCOMB][NPAD];   // 8 KB: [Wih^T ; 0 ; Whh^T ; 0]
    __shared__ float biasv[NPAD];
    __shared__ float axh[BATCH][KCOMB];    // 8 KB: per-row [x(15),0, h(15),0]
    __shared__ float gbuf[BATCH][NPAD];    // 16 KB: gate pre-activations
    __shared__ float cbuf[BATCH][16];      // 4 KB: cell state

    const int dir  = blockIdx.x;           // 0 = forward, 1 = backward
    const int tid  = threadIdx.x;
    const int lane = tid & 31;
    const int wave = tid >> 5;

    const float* w_ih = dir ? w_ih_b : w_ih_f;
    const float* w_hh = dir ? w_hh_b : w_hh_f;
    const float* bias = dir ? b_b    : b_f;

    // ---- stage weights / bias / initial state -----------------------------
    for (int e = tid; e < KCOMB * NPAD; e += 256) {
        int k = e >> 6, n = e & 63;
        float v = 0.0f;
        if (n < GDIM) {
            if (k < EDIM)                      v = w_ih[n * EDIM + k];
            else if (k >= 16 && k < 16 + HDIM) v = w_hh[n * HDIM + (k - 16)];
        }
        wcomb[k][n] = v;
    }
    for (int n = tid; n < NPAD; n += 256) biasv[n] = (n < GDIM) ? bias[n] : 0.0f;
    for (int e = tid; e < BATCH * 16; e += 256) {
        int r = e >> 4, j = e & 15;
        float hv = (j < HDIM) ? h0[dir * BATCH * HDIM + r * HDIM + j] : 0.0f;
        float cv = (j < HDIM) ? c0[dir * BATCH * HDIM + r * HDIM + j] : 0.0f;
        axh[r][16 + j] = hv;
        cbuf[r][j]     = cv;
    }

    // ---- sequential scan --------------------------------------------------
    for (int s = 0; s < TMAX; ++s) {
        // Stage x_t (embedding gather); backward reads reversed-in-length.
        for (int e = tid; e < BATCH * 16; e += 256) {
            int r = e >> 4, j = e & 15;
            int t = s;
            if (dir) { t = lengths[r] - 1 - s; if (t < 0) t = 0; }
            int tok = data[r * TMAX + t];
            axh[r][j] = (j < EDIM) ? emb[(size_t)tok * EDIM + j] : 0.0f;
            if (j == 0) {   // prefetch next step's embedding row into caches
                int tn = dir ? (t - 1) : (t + 1);
                if (tn >= 0 && tn < TMAX) {
                    int tok2 = data[r * TMAX + tn];
                    __builtin_prefetch(emb + (size_t)tok2 * EDIM, 0, 1);
                }
            }
        }
        __syncthreads();

        // gbuf = axh @ wcomb + bias   ([64x32]@[32x64], f32 WMMA 16x16x4)
        for (int tile = wave; tile < 16; tile += 8) {
            int mt = tile >> 2, nt = tile & 3;
            int mrow  = (mt << 4) + (lane & 15);
            int ncol  = (nt << 4) + (lane & 15);
            int khalf = (lane >> 4) << 1;
            float bv = biasv[ncol];
            v8f acc = {bv, bv, bv, bv, bv, bv, bv, bv};
#pragma unroll
            for (int kk = 0; kk < 8; ++kk) {
                int k = (kk << 2) + khalf;
                v2f a; a.x = axh[mrow][k];   a.y = axh[mrow][k + 1];
                v2f b; b.x = wcomb[k][ncol]; b.y = wcomb[k + 1][ncol];
                acc = __builtin_amdgcn_wmma_f32_16x16x4_f32(
                          false, a, false, b, (short)0, acc, false, false);
            }
            int mbase = (mt << 4) + ((lane >> 4) << 3);
#pragma unroll
            for (int r2 = 0; r2 < 8; ++r2)
                gbuf[mbase + r2][ncol] = acc[r2];
        }
        __syncthreads();

        // Gates (torch order i,f,g,o), state update, emit h.
        for (int e = tid; e < BATCH * HDIM; e += 256) {
            int r = e / HDIM, j = e - r * HDIM;
            float gi = gbuf[r][j];
            float gf = gbuf[r][HDIM + j];
            float gg = gbuf[r][2 * HDIM + j];
            float go = gbuf[r][3 * HDIM + j];
            float c  = sigm_(gf) * cbuf[r][j] + sigm_(gi) * tanhf(gg);
            float h  = sigm_(go) * tanhf(c);
            cbuf[r][j]     = c;
            axh[r][16 + j] = h;
            int tpos = dir ? (lengths[r] - 1 - s) : s;
            if (tpos >= 0)
                Hcat[((size_t)tpos * BATCH + r) * KCOMB + dir * HDIM + j] = h;
        }
        __syncthreads();
    }
}

__global__ __launch_bounds__(128)
void linear_kernel(const float* __restrict__ Hcat,     // (T*B, 32)
                   const int* __restrict__ lengths,    // (B)
                   const float* __restrict__ lin_w,    // (45, 30)
                   const float* __restrict__ lin_b,    // (45)
                   float* __restrict__ out)            // (B, T, 45)
{
    __shared__ float hs[16][KCOMB];   // 2 KB
    __shared__ float lw[KCOMB][48];   // 6 KB
    __shared__ float lb[48];

    const int tid  = threadIdx.x;
    const int lane = tid & 31;
    const int wave = tid >> 5;
    const int rowbase = blockIdx.x << 4;   // 16 rows (t*B+b flattened)

    for (int e = tid; e < KCOMB * 48; e += 128) {
        int k = e / 48, n = e - k * 48;
        lw[k][n] = (k < 2 * HDIM && n < 45) ? lin_w[n * (2 * HDIM) + k] : 0.0f;
    }
    for (int n = tid; n < 48; n += 128) lb[n] = (n < 45) ? lin_b[n] : 0.0f;
    // Stage rows; zero invalid (t >= length) rows and padding columns so
    // masked positions produce exactly bias (matches reference semantics).
    for (int e = tid; e < 16 * KCOMB; e += 128) {
        int r = e >> 5, k = e & 31;
        int row = rowbase + r;
        int t = row >> 6, b = row & 63;
        float v = 0.0f;
        if (k < 2 * HDIM && t < lengths[b]) v = Hcat[(size_t)row * KCOMB + k];
        hs[r][k] = v;
    }
    __syncthreads();

    if (wave < 3) {                        // 3 N-tiles cover 45 (pad 48)
        int ncol  = (wave << 4) + (lane & 15);
        int khalf = (lane >> 4) << 1;
        float bv = lb[ncol];
        v8f acc = {bv, bv, bv, bv, bv, bv, bv, bv};
#pragma unroll
        for (int kk = 0; kk < 8; ++kk) {
            int k = (kk << 2) + khalf;
            v2f a; a.x = hs[lane & 15][k]; a.y = hs[lane & 15][k + 1];
            v2f b; b.x = lw[k][ncol];      b.y = lw[k + 1][ncol];
            acc = __builtin_amdgcn_wmma_f32_16x16x4_f32(
                      false, a, false, b, (short)0, acc, false, false);
        }
        if (ncol < 45) {
            int mbase = (lane >> 4) << 3;
#pragma unroll
            for (int r2 = 0; r2 < 8; ++r2) {
                int row = rowbase + mbase + r2;
                int t = row >> 6, b = row & 63;
                out[((size_t)b * TMAX + t) * 45 + ncol] = acc[r2];
            }
        }
    }
}

extern "C" void kernel_launch(void* const* d_in, const int* in_sizes, int n_in,
                              void* d_out, int out_size, void* d_ws, size_t ws_size,
                              hipStream_t stream) {
    (void)in_sizes; (void)n_in; (void)out_size; (void)ws_size;
    const int*   data    = (const int*)  d_in[0];
    // d_in[1] = innerSize scalar (== TMAX, hardcoded)
    const int*   lengths = (const int*)  d_in[2];
    const float* emb     = (const float*)d_in[3];
    const float* h0      = (const float*)d_in[4];
    const float* c0      = (const float*)d_in[5];
    const float* w_ih_f  = (const float*)d_in[6];
    const float* w_hh_f  = (const float*)d_in[7];
    const float* b_f     = (const float*)d_in[8];
    const float* w_ih_b  = (const float*)d_in[9];
    const float* w_hh_b  = (const float*)d_in[10];
    const float* b_b     = (const float*)d_in[11];
    const float* lin_w   = (const float*)d_in[12];
    const float* lin_b   = (const float*)d_in[13];

    float* Hcat = (float*)d_ws;                 // (T*B) x 32 floats = 32 MB
    float* out  = (float*)d_out;                // (B, T, 45) float32

    lstm_seq_kernel<<<2, 256, 0, stream>>>(data, lengths, emb, h0, c0,
                                           w_ih_f, w_hh_f, b_f,
                                           w_ih_b, w_hh_b, b_b, Hcat);
    linear_kernel<<<(TMAX * BATCH) / 16, 128, 0, stream>>>(Hcat, lengths,
                                                           lin_w, lin_b, out);
}